// Blur_38405597560893
// MI455X (gfx1250) — compile-verified
//
#include <hip/hip_runtime.h>
#include <stdint.h>

#ifndef __has_builtin
#define __has_builtin(x) 0
#endif

#if defined(__HIP_DEVICE_COMPILE__) && \
    __has_builtin(__builtin_amdgcn_global_load_async_to_lds_b32) && \
    __has_builtin(__builtin_amdgcn_s_wait_asynccnt)
#define USE_ASYNC_LDS 1
#else
#define USE_ASYNC_LDS 0
#endif

namespace {

constexpr int  kN  = 4;
constexpr int  kC  = 3;
constexpr int  kS  = 512;
constexpr int  kD  = 17;
constexpr int  BLK = 256;              // 8 wave32 waves per block
constexpr long kSS = (long)kS * kS;

#if USE_ASYNC_LDS
typedef __attribute__((address_space(1))) int* gptr_b32;
typedef __attribute__((address_space(3))) int* lptr_b32;
#endif

__device__ __forceinline__ float clampf(float v, float lo, float hi) {
    return fminf(fmaxf(v, lo), hi);
}

// Bilinear sample of a [C=3,S,S] image at (ix, iy), grid_sample semantics
// (padding_mode='zeros'); ix/iy are pre-clamped to [-0.5, S-0.5].
__device__ __forceinline__ void sample3(const float* __restrict__ img,
                                        float ix, float iy,
                                        float& v0, float& v1, float& v2) {
    const float x0f = floorf(ix), y0f = floorf(iy);
    float wx1 = ix - x0f, wy1 = iy - y0f;
    float wx0 = 1.0f - wx1, wy0 = 1.0f - wy1;
    const int x0 = (int)x0f, y0 = (int)y0f;      // x0,y0 in [-1, S-1]
    const int x1 = x0 + 1,   y1 = y0 + 1;        // x1,y1 in [ 0, S  ]
    // fold out-of-bounds "zeros" padding into the weights
    wx0 = (x0 >= 0)      ? wx0 : 0.0f;
    wx1 = (x1 <= kS - 1) ? wx1 : 0.0f;
    wy0 = (y0 >= 0)      ? wy0 : 0.0f;
    wy1 = (y1 <= kS - 1) ? wy1 : 0.0f;
    const int xi0 = (x0 > 0) ? x0 : 0;
    const int xi1 = (x1 < kS - 1) ? x1 : (kS - 1);
    const int yi0 = (y0 > 0) ? y0 : 0;
    const int yi1 = (y1 < kS - 1) ? y1 : (kS - 1);

    const float w00 = wx0 * wy0, w01 = wx1 * wy0;
    const float w10 = wx0 * wy1, w11 = wx1 * wy1;

    const float* r00 = img + (long)yi0 * kS + xi0;
    const float* r01 = img + (long)yi0 * kS + xi1;
    const float* r10 = img + (long)yi1 * kS + xi0;
    const float* r11 = img + (long)yi1 * kS + xi1;

    // channel stride is a compile-time immediate (1 MB / 2 MB, fits IOFFSET)
    v0 = r00[0]        * w00 + r01[0]        * w01 + r10[0]        * w10 + r11[0]        * w11;
    v1 = r00[kSS]      * w00 + r01[kSS]      * w01 + r10[kSS]      * w10 + r11[kSS]      * w11;
    v2 = r00[2 * kSS]  * w00 + r01[2 * kSS]  * w01 + r10[2 * kSS]  * w10 + r11[2 * kSS]  * w11;
}

__global__ __launch_bounds__(BLK) void blur_flow_kernel(
    const float* __restrict__ X0, const float* __restrict__ X1,
    const float* __restrict__ Wt, const float* __restrict__ sx,
    const float* __restrict__ sy, const float* __restrict__ flow,
    float* __restrict__ out)
{
#if USE_ASYNC_LDS
    // Per-thread private 16B slot, double buffered: no cross-thread deps,
    // hence NO workgroup barriers anywhere in the kernel.
    __shared__ __align__(16) float smem[2][BLK][4];
#endif
    const int tid = (int)threadIdx.x;
    int b = (int)blockIdx.x;
    const int xt = b & 1;  b >>= 1;          // 2 tiles of 256px per row
    const int y  = b & (kS - 1);
    const int n  = b >> 9;                   // S = 512 rows per batch
    const int x  = xt * BLK + tid;

    const float2 fl = ((const float2*)flow)[((long)n * kS + y) * kS + x];
    const float fx = fl.x, fy = fl.y;

    const float* imgA = X0 + (long)n * kC * kSS;
    const float* imgB = X1 + (long)n * kC * kSS;

    const long rowoff = (long)y * kS + (long)xt * BLK;     // tile base inside a plane
    const float* sxr  = sx + (long)n * kD * kSS + rowoff + tid;        // + d*kSS
    const float* syr  = sy + (long)n * kD * kSS + rowoff + tid;        // + d*kSS
    const float* w01r = Wt + (long)n * (2 * kD) * kSS + rowoff + tid;  // + d*kSS
    const float* w10r = w01r + (long)kD * kSS;                         // + d*kSS

#if USE_ASYNC_LDS
    // 4 coalesced async b32 copies per lane into the lane's own LDS slot.
    auto stage = [&](int d, int buf) {
        const long off = (long)d * kSS;
        float* slot = &smem[buf][tid][0];
        __builtin_amdgcn_global_load_async_to_lds_b32(
            (gptr_b32)(uintptr_t)(sxr + off),  (lptr_b32)(uintptr_t)(slot + 0), 0, 0);
        __builtin_amdgcn_global_load_async_to_lds_b32(
            (gptr_b32)(uintptr_t)(syr + off),  (lptr_b32)(uintptr_t)(slot + 1), 0, 0);
        __builtin_amdgcn_global_load_async_to_lds_b32(
            (gptr_b32)(uintptr_t)(w01r + off), (lptr_b32)(uintptr_t)(slot + 2), 0, 0);
        __builtin_amdgcn_global_load_async_to_lds_b32(
            (gptr_b32)(uintptr_t)(w10r + off), (lptr_b32)(uintptr_t)(slot + 3), 0, 0);
    };
    stage(0, 0);
#endif

    float csx = 0.0f, csy = 0.0f;
    float acc0 = 0.0f, acc1 = 0.0f, acc2 = 0.0f;
    const float xb = (float)x - 0.5f;       // grid algebra: ix = clamp(xb - csx*fx)
    const float yb = (float)y - 0.5f;

    for (int d = 0; d < kD; ++d) {
#if USE_ASYNC_LDS
        const int buf = d & 1;
        if (d + 1 < kD) {
            stage(d + 1, buf ^ 1);
            __builtin_amdgcn_s_wait_asynccnt(4);   // this lane's 4 loads for d done (in-order)
        } else {
            __builtin_amdgcn_s_wait_asynccnt(0);
        }
        const float4 sv = *(const float4*)&smem[buf][tid][0];   // one ds_load_b128
        const float sxv = sv.x, syv = sv.y, w01 = sv.z, w10 = sv.w;
#else
        const float sxv = __builtin_nontemporal_load(sxr  + (long)d * kSS);
        const float syv = __builtin_nontemporal_load(syr  + (long)d * kSS);
        const float w01 = __builtin_nontemporal_load(w01r + (long)d * kSS);
        const float w10 = __builtin_nontemporal_load(w10r + (long)d * kSS);
        if (d + 2 < kD) {   // keep the stream ahead: global_prefetch_b8
            __builtin_prefetch(sxr  + (long)(d + 2) * kSS, 0, 0);
            __builtin_prefetch(syr  + (long)(d + 2) * kSS, 0, 0);
            __builtin_prefetch(w01r + (long)(d + 2) * kSS, 0, 0);
            __builtin_prefetch(w10r + (long)(d + 2) * kSS, 0, 0);
        }
#endif

        csx += sxv;  csy += syv;
        const float tx = csx * fx, ty = csy * fy;

        // forward warp samples X0, backward warp (theta - flow) samples X1
        const float ixA = clampf(xb - tx,      -0.5f, (float)kS - 0.5f);
        const float iyA = clampf(yb - ty,      -0.5f, (float)kS - 0.5f);
        const float ixB = clampf(xb - tx + fx, -0.5f, (float)kS - 0.5f);
        const float iyB = clampf(yb - ty + fy, -0.5f, (float)kS - 0.5f);

        float a0, a1, a2, b0, b1, b2;
        sample3(imgA, ixA, iyA, a0, a1, a2);
        sample3(imgB, ixB, iyB, b0, b1, b2);

        acc0 += clampf(a0, 0.0f, 255.0f) * w01 + clampf(b0, 0.0f, 255.0f) * w10;
        acc1 += clampf(a1, 0.0f, 255.0f) * w01 + clampf(b1, 0.0f, 255.0f) * w10;
        acc2 += clampf(a2, 0.0f, 255.0f) * w01 + clampf(b2, 0.0f, 255.0f) * w10;
    }

    const float inv = 1.0f / (float)(2 * kD);
    float* op = out + (long)n * kC * kSS + rowoff + tid;
    __builtin_nontemporal_store(clampf(acc0 * inv, 0.0f, 255.0f), op);
    __builtin_nontemporal_store(clampf(acc1 * inv, 0.0f, 255.0f), op + kSS);
    __builtin_nontemporal_store(clampf(acc2 * inv, 0.0f, 255.0f), op + 2 * kSS);
}

}  // namespace

extern "C" void kernel_launch(void* const* d_in, const int* in_sizes, int n_in,
                              void* d_out, int out_size, void* d_ws, size_t ws_size,
                              hipStream_t stream) {
    (void)in_sizes; (void)n_in; (void)out_size; (void)d_ws; (void)ws_size;
    const float* X0   = (const float*)d_in[0];
    const float* X1   = (const float*)d_in[1];
    const float* Wt   = (const float*)d_in[2];
    const float* sx   = (const float*)d_in[3];
    const float* sy   = (const float*)d_in[4];
    const float* flow = (const float*)d_in[5];
    float* out = (float*)d_out;

    const int blocks = kN * kS * (kS / BLK);   // 4096 blocks x 256 threads
    blur_flow_kernel<<<blocks, BLK, 0, stream>>>(X0, X1, Wt, sx, sy, flow, out);
}